// AuxiliaryLoss_43370579755543
// MI455X (gfx1250) — compile-verified
//
#include <hip/hip_runtime.h>
#include <math.h>

typedef __attribute__((ext_vector_type(2))) float v2f;
typedef __attribute__((ext_vector_type(8))) float v8f;

#define KDIM 32   // padded K (28 real features + 4 zeros)

__constant__ float c_chi[21][4] = {
    {0,0,0,0},{1,1,1,1},{1,1,0,0},{1,1,0,0},{1,0,0,0},{1,1,1,0},{1,1,1,0},
    {0,0,0,0},{1,1,0,0},{1,1,0,0},{1,1,0,0},{1,1,1,1},{1,1,1,0},{1,1,0,0},
    {1,1,0,0},{1,0,0,0},{1,0,0,0},{1,1,0,0},{1,1,0,0},{1,0,0,0},{0,0,0,0}};

__global__ void k_zero(float* __restrict__ p, int n) {
    int i = blockIdx.x * blockDim.x + threadIdx.x;
    if (i < n) p[i] = 0.f;
}

// Fragment-major packing: feature k (0..31) for element idx is stored at
//   out[idx*32 + hi*16 + ks*2 + c]   where k = 4*ks + 2*hi + c.
// With this layout, the 16 floats a lane (lo,hi) needs for ALL 8 K-steps of a
// WMMA fragment are contiguous and 64B aligned -> b128 loads in the GEMM.
__device__ __forceinline__ void store_packed(float* __restrict__ o,
                                             const float* __restrict__ f) {
#pragma unroll
    for (int hi = 0; hi < 2; hi++)
#pragma unroll
        for (int ks = 0; ks < 8; ks++)
#pragma unroll
            for (int c = 0; c < 2; c++)
                o[hi * 16 + ks * 2 + c] = f[4 * ks + 2 * hi + c];
}

// Row features m_i: per (l,b,i), 32 floats.
// A_i = [Rp_i^T | -Rt_i^T] (3x6); b_i = -A[:, :3]*tp - A[:, 3:]*tt
// m = [diag(A^T A) (6), 2*offdiag p<q (15), 2*A^T b (6), b.b (1), pad (4)]
__global__ void k_rowfeat(const float* __restrict__ Rp_all, const float* __restrict__ tp_all,
                          const float* __restrict__ Rt_all, const float* __restrict__ tt_all,
                          float* __restrict__ rowf, int LBN, int B, int N) {
    int idx = blockIdx.x * blockDim.x + threadIdx.x;
    if (idx >= LBN) return;
    int i = idx % N;
    int b = (idx / N) % B;
    const float* Rp = Rp_all + (size_t)idx * 9;
    const float* tp = tp_all + (size_t)idx * 3;
    const float* Rt = Rt_all + (size_t)(b * N + i) * 9;
    const float* tt = tt_all + (size_t)(b * N + i) * 3;

    float A[3][6];
#pragma unroll
    for (int r = 0; r < 3; r++)
#pragma unroll
        for (int c = 0; c < 3; c++) {
            A[r][c]     =  Rp[c * 3 + r];   // Rp^T
            A[r][c + 3] = -Rt[c * 3 + r];   // -Rt^T
        }
    float bv[3];
#pragma unroll
    for (int r = 0; r < 3; r++)
        bv[r] = -(A[r][0]*tp[0] + A[r][1]*tp[1] + A[r][2]*tp[2])
                -(A[r][3]*tt[0] + A[r][4]*tt[1] + A[r][5]*tt[2]);

    float f[KDIM];
#pragma unroll
    for (int p = 0; p < 6; p++) {
        float s = 0.f;
#pragma unroll
        for (int r = 0; r < 3; r++) s += A[r][p] * A[r][p];
        f[p] = s;
    }
    int k = 6;
#pragma unroll
    for (int p = 0; p < 6; p++)
#pragma unroll
        for (int q = p + 1; q < 6; q++) {
            float s = 0.f;
#pragma unroll
            for (int r = 0; r < 3; r++) s += A[r][p] * A[r][q];
            f[k++] = 2.f * s;
        }
#pragma unroll
    for (int p = 0; p < 6; p++) {
        float s = 0.f;
#pragma unroll
        for (int r = 0; r < 3; r++) s += A[r][p] * bv[r];
        f[21 + p] = 2.f * s;
    }
    f[27] = bv[0]*bv[0] + bv[1]*bv[1] + bv[2]*bv[2];
    f[28] = 0.f; f[29] = 0.f; f[30] = 0.f; f[31] = 0.f;
    store_packed(rowf + (size_t)idx * KDIM, f);
}

// Column features w_j: per (l,b,j), u = [tp_j; tt_j]
// w = [u_p^2 (6), u_p*u_q p<q (15), u (6), 1, pad (4)]  (same pair order as rows)
__global__ void k_colfeat(const float* __restrict__ tp_all, const float* __restrict__ tt_all,
                          float* __restrict__ colf, int LBN, int B, int N) {
    int idx = blockIdx.x * blockDim.x + threadIdx.x;
    if (idx >= LBN) return;
    int j = idx % N;
    int b = (idx / N) % B;
    const float* tp = tp_all + (size_t)idx * 3;
    const float* tt = tt_all + (size_t)(b * N + j) * 3;
    float u[6] = {tp[0], tp[1], tp[2], tt[0], tt[1], tt[2]};
    float f[KDIM];
#pragma unroll
    for (int p = 0; p < 6; p++) f[p] = u[p] * u[p];
    int k = 6;
#pragma unroll
    for (int p = 0; p < 6; p++)
#pragma unroll
        for (int q = p + 1; q < 6; q++) f[k++] = u[p] * u[q];
#pragma unroll
    for (int p = 0; p < 6; p++) f[21 + p] = u[p];
    f[27] = 1.f;
    f[28] = 0.f; f[29] = 0.f; f[30] = 0.f; f[31] = 0.f;
    store_packed(colf + (size_t)idx * KDIM, f);
}

// FAPE GEMM: one wave computes a 16x64 strip of D[i,j] = m_i . w_j via
// v_wmma_f32_16x16x4_f32. All fragments are loaded into registers up front
// (contiguous b128-able loads thanks to fragment-major packing), then the
// 32-WMMA chain runs with no memory ops inside; sqrt/clamp/mask/reduce is
// fused in the C fragments.
__global__ void __launch_bounds__(256)
k_fape(const float* __restrict__ rowf, const float* __restrict__ colf,
       const float* __restrict__ seq, float* __restrict__ fape_sum,
       int LB, int B, int N) {
    const int lane = threadIdx.x & 31;
    const int wave = blockIdx.x * (blockDim.x >> 5) + (threadIdx.x >> 5);
    const int jblocks = N >> 6;                 // strips of 4 tiles along j
    const int strips_per_lb = (N >> 4) * jblocks;
    const int lb = wave / strips_per_lb;
    if (lb >= LB) return;                       // wave-uniform: EXEC stays all-1s
    const int s  = wave % strips_per_lb;
    const int i0 = (s / jblocks) << 4;
    const int j0 = (s % jblocks) << 6;
    const int b  = lb % B;

    const float* rf = rowf + (size_t)lb * N * KDIM;
    const float* cf = colf + (size_t)lb * N * KDIM;
    const int lo = lane & 15, hi = lane >> 4;

    // ---- issue ALL fragment loads first (contiguous 16-float blocks) ----
    const float* ab = rf + (size_t)(i0 + lo) * KDIM + hi * 16;
    v2f afrag[8];
#pragma unroll
    for (int q = 0; q < 8; q++) afrag[q] = *(const v2f*)(ab + 2 * q);

    v2f bfrag[4][8];
#pragma unroll
    for (int t = 0; t < 4; t++) {
        const float* bb = cf + (size_t)(j0 + t * 16 + lo) * KDIM + hi * 16;
#pragma unroll
        for (int q = 0; q < 8; q++) bfrag[t][q] = *(const v2f*)(bb + 2 * q);
    }

    const float* sm = seq + (size_t)b * N;
    float rmask[8];
#pragma unroll
    for (int r = 0; r < 8; r++) rmask[r] = sm[i0 + r + 8 * hi];
    float jmask[4];
#pragma unroll
    for (int t = 0; t < 4; t++) jmask[t] = sm[j0 + t * 16 + lo];

    // ---- pure WMMA chain: 8 K-steps x 4 independent accumulators ----
    v8f acc[4];
#pragma unroll
    for (int t = 0; t < 4; t++) acc[t] = (v8f){0.f,0.f,0.f,0.f,0.f,0.f,0.f,0.f};
#pragma unroll
    for (int q = 0; q < 8; q++) {
#pragma unroll
        for (int t = 0; t < 4; t++) {
            acc[t] = __builtin_amdgcn_wmma_f32_16x16x4_f32(
                false, afrag[q], false, bfrag[t][q], (short)0, acc[t], false, false);
        }
    }

    // ---- epilogue: C layout VGPR r -> row i0 + r + 8*hi, col j0 + 16*t + lo ----
    float lsum = 0.f;
#pragma unroll
    for (int t = 0; t < 4; t++) {
#pragma unroll
        for (int r = 0; r < 8; r++) {
            float d2 = acc[t][r];
            float dist = sqrtf(fmaxf(d2, 0.f) + 1e-4f);
            dist = fminf(dist, 10.f);
            lsum += dist * rmask[r] * jmask[t];
        }
    }
#pragma unroll
    for (int off = 16; off > 0; off >>= 1) lsum += __shfl_xor(lsum, off, 32);
    if (lane == 0) atomicAdd(&fape_sum[lb], lsum);
}

__global__ void __launch_bounds__(256)
k_torsion(const float* __restrict__ traj_tor, const float* __restrict__ true_tor,
          const float* __restrict__ alt_tor, const int* __restrict__ res,
          const float* __restrict__ seq, float* __restrict__ tsum,
          float* __restrict__ asum, int B, int N) {
    int idx = blockIdx.x * blockDim.x + threadIdx.x;   // (l*B+b)*N + n
    int n  = idx % N;
    int lb = idx / N;
    int b  = lb % B;
    const float* pt = traj_tor + (size_t)idx * 14;
    const float* tt = true_tor + (size_t)(b * N + n) * 14;
    const float* at = alt_tor  + (size_t)(b * N + n) * 14;
    int rt = res[b * N + n];
    rt = rt < 0 ? 0 : (rt > 20 ? 20 : rt);
    float s = seq[b * N + n];
    float tl = 0.f, al = 0.f;
#pragma unroll
    for (int t = 0; t < 7; t++) {
        float x = pt[2*t], y = pt[2*t+1];
        float nrm = sqrtf(x*x + y*y + 1e-8f);
        float ux = x / nrm, uy = y / nrm;
        float dx = tt[2*t] - ux, dy = tt[2*t+1] - uy;
        float d1 = dx*dx + dy*dy;
        dx = at[2*t] - ux; dy = at[2*t+1] - uy;
        float d2 = dx*dx + dy*dy;
        float dsq = fminf(d1, d2);
        float m = (t < 3 ? 1.f : c_chi[rt][t-3]) * s;
        tl += dsq * m;
        al += fabsf(nrm - 1.f) * m;
    }
    __shared__ float sh[2][8];
#pragma unroll
    for (int off = 16; off > 0; off >>= 1) {
        tl += __shfl_xor(tl, off, 32);
        al += __shfl_xor(al, off, 32);
    }
    int w = threadIdx.x >> 5, ln = threadIdx.x & 31;
    if (ln == 0) { sh[0][w] = tl; sh[1][w] = al; }
    __syncthreads();
    if (threadIdx.x == 0) {
        float t0 = 0.f, a0 = 0.f;
        for (int w2 = 0; w2 < (int)(blockDim.x >> 5); w2++) { t0 += sh[0][w2]; a0 += sh[1][w2]; }
        atomicAdd(&tsum[lb], t0);
        atomicAdd(&asum[lb], a0);
    }
}

__global__ void __launch_bounds__(256)
k_masks(const int* __restrict__ res, const float* __restrict__ seq,
        float* __restrict__ ssum, float* __restrict__ tmn, int N) {
    int idx = blockIdx.x * blockDim.x + threadIdx.x;   // b*N + n
    int b = idx / N;
    float s = seq[idx];
    int rt = res[idx];
    rt = rt < 0 ? 0 : (rt > 20 ? 20 : rt);
    float tm = s * (3.f + c_chi[rt][0] + c_chi[rt][1] + c_chi[rt][2] + c_chi[rt][3]);
    __shared__ float sh[2][8];
#pragma unroll
    for (int off = 16; off > 0; off >>= 1) {
        s  += __shfl_xor(s, off, 32);
        tm += __shfl_xor(tm, off, 32);
    }
    int w = threadIdx.x >> 5, ln = threadIdx.x & 31;
    if (ln == 0) { sh[0][w] = s; sh[1][w] = tm; }
    __syncthreads();
    if (threadIdx.x == 0) {
        float s0 = 0.f, t0 = 0.f;
        for (int w2 = 0; w2 < (int)(blockDim.x >> 5); w2++) { s0 += sh[0][w2]; t0 += sh[1][w2]; }
        atomicAdd(&ssum[b], s0);
        atomicAdd(&tmn[b], t0);
    }
}

__global__ void k_finalize(const float* __restrict__ fape, const float* __restrict__ tsum,
                           const float* __restrict__ asum, const float* __restrict__ ssum,
                           const float* __restrict__ tmn, float* __restrict__ out,
                           int L, int B) {
    int b = threadIdx.x;
    if (b >= B) return;
    float pc = fmaxf(ssum[b] * ssum[b], 1.f);   // pair_count = (sum seq)^2
    float nz = fmaxf(tmn[b], 1.f);
    float acc = 0.f;
    for (int l = 0; l < L; l++) {
        acc += fape[l * B + b] / (10.f * pc)
             + (tsum[l * B + b] + 0.02f * asum[l * B + b]) / nz;
    }
    out[b] = acc / (float)L;
}

extern "C" void kernel_launch(void* const* d_in, const int* in_sizes, int n_in,
                              void* d_out, int out_size, void* d_ws, size_t ws_size,
                              hipStream_t stream) {
    const float* traj_rot   = (const float*)d_in[0];
    const float* traj_trans = (const float*)d_in[1];
    const float* traj_tor   = (const float*)d_in[2];
    const float* true_rot   = (const float*)d_in[3];
    const float* true_trans = (const float*)d_in[4];
    const float* true_tor   = (const float*)d_in[5];
    const float* alt_tor    = (const float*)d_in[6];
    const int*   res_types  = (const int*)d_in[7];
    const float* seq_mask   = (const float*)d_in[8];

    const int B   = out_size;                 // 4
    const int BN  = in_sizes[8];              // B*N
    const int N   = BN / B;                   // 1024
    const int LBN = in_sizes[1] / 3;          // L*B*N
    const int L   = LBN / BN;                 // 8
    const int LB  = L * B;

    float* ws       = (float*)d_ws;
    float* rowf     = ws;                             // LBN*32 floats (packed)
    float* colf     = rowf + (size_t)LBN * KDIM;      // LBN*32 floats (packed)
    float* fape_sum = colf + (size_t)LBN * KDIM;      // LB
    float* tsum     = fape_sum + LB;                  // LB
    float* asum     = tsum + LB;                      // LB
    float* ssum     = asum + LB;                      // B
    float* tmn      = ssum + B;                       // B
    int nacc = 3 * LB + 2 * B;

    k_zero<<<(nacc + 255) / 256, 256, 0, stream>>>(fape_sum, nacc);
    k_rowfeat<<<(LBN + 255) / 256, 256, 0, stream>>>(
        traj_rot, traj_trans, true_rot, true_trans, rowf, LBN, B, N);
    k_colfeat<<<(LBN + 255) / 256, 256, 0, stream>>>(
        traj_trans, true_trans, colf, LBN, B, N);

    int strips = LB * (N >> 4) * (N >> 6);            // 16-row x 64-col strips
    k_fape<<<(strips + 7) / 8, 256, 0, stream>>>(rowf, colf, seq_mask, fape_sum, LB, B, N);

    k_torsion<<<LBN / 256, 256, 0, stream>>>(
        traj_tor, true_tor, alt_tor, res_types, seq_mask, tsum, asum, B, N);
    k_masks<<<BN / 256, 256, 0, stream>>>(res_types, seq_mask, ssum, tmn, N);
    k_finalize<<<1, 32, 0, stream>>>(fape_sum, tsum, asum, ssum, tmn, (float*)d_out, L, B);
}